// GatedShortBlock_16346645528822
// MI455X (gfx1250) — compile-verified
//
#include <hip/hip_runtime.h>

// GatedShortBlock for MI455X (gfx1250, wave32).
// out = (Cg * depthwise_causal_conv(Bg*Xg)) @ w2^T, where [Bg|Cg|Xg] = x @ w1^T.
// fp32 throughout -> V_WMMA_F32_16X16X4_F32 matrix path.

#define D_MODEL 2048
#define SEQ     4096
#define BATCH   4
#define BS      (BATCH * SEQ)     // 16384 rows
#define KTILE   32                // K-chunk staged in LDS
#define KP      36                // padded LDS row stride (dwords), even for b64 frags
#define MTILE   128
#define NTILE   128

typedef float v2f __attribute__((ext_vector_type(2)));
typedef float v4f __attribute__((ext_vector_type(4)));
typedef float v8f __attribute__((ext_vector_type(8)));

// ---------------------------------------------------------------------------
// Cooperative 128x32 fp32 tile load: global (row stride D_MODEL) -> LDS (stride KP).
// 256 threads: each thread copies a 16-float half-row via 4x b128.
// ---------------------------------------------------------------------------
__device__ __forceinline__ void load_tile_128x32(const float* __restrict__ g,
                                                 float* lds, int t) {
    const int row = t >> 1;
    const int col = (t & 1) << 4;
    const v4f* __restrict__ src = (const v4f*)(g + (size_t)row * D_MODEL + col);
    v4f* dst = (v4f*)(lds + row * KP + col);
    dst[0] = src[0];
    dst[1] = src[1];
    dst[2] = src[2];
    dst[3] = src[3];
    // Speculative prefetch of next K-tile (global_prefetch_b8); dropped if OOB.
    __builtin_prefetch((const void*)(g + (size_t)row * D_MODEL + col + KTILE), 0, 0);
}

// ---------------------------------------------------------------------------
// One KTILE (=32) slab of WMMA work for a wave's 32x64 output block.
// A fragment: a[j] = A[m = 16i + ln][k = 2*kh + j]   (16x4, 2 VGPRs/lane)
// B fragment: b[j] = B[k = 2*kh + j][n = 16j' + ln]  (4x16, 2 VGPRs/lane)
// Both are contiguous-in-k in LDS -> aligned ds_load_b64.
// ---------------------------------------------------------------------------
__device__ __forceinline__ void wmma_ktile(const float* As, const float* Ws,
                                           v8f acc[2][4],
                                           int wm, int wn, int ln, int kh) {
#pragma unroll
    for (int ks = 0; ks < KTILE / 4; ++ks) {
        const int kk = ks * 4 + kh * 2;
        v2f a[2], b[4];
#pragma unroll
        for (int i = 0; i < 2; ++i)
            a[i] = *(const v2f*)(As + (wm * 32 + i * 16 + ln) * KP + kk);
#pragma unroll
        for (int j = 0; j < 4; ++j)
            b[j] = *(const v2f*)(Ws + (wn * 64 + j * 16 + ln) * KP + kk);
#pragma unroll
        for (int i = 0; i < 2; ++i)
#pragma unroll
            for (int j = 0; j < 4; ++j)
                acc[i][j] = __builtin_amdgcn_wmma_f32_16x16x4_f32(
                    false, a[i], false, b[j], (short)0, acc[i][j], false, false);
    }
}

// C/D 16x16 layout: VGPR r -> M = r + 8*kh, lane ln -> N = ln.
__device__ __forceinline__ void store_acc(float* __restrict__ dst,
                                          const v8f acc[2][4],
                                          int m0, int n0,
                                          int wm, int wn, int ln, int kh) {
#pragma unroll
    for (int i = 0; i < 2; ++i)
#pragma unroll
        for (int j = 0; j < 4; ++j)
#pragma unroll
            for (int r = 0; r < 8; ++r) {
                const int row = m0 + wm * 32 + i * 16 + kh * 8 + r;
                const int col = n0 + wn * 64 + j * 16 + ln;
                dst[(size_t)row * D_MODEL + col] = acc[i][j][r];
            }
}

// ---------------------------------------------------------------------------
// Kernel 1: per WG, a 128(rows) x 128(d) tile.
// Pass A (single K loop): Bg and Xg accumulators together -> gated = Bg*Xg.
// Pass B: Cg accumulator -> cg buffer.
// ---------------------------------------------------------------------------
__global__ __launch_bounds__(256)
void k1_gemm_gate(const float* __restrict__ x, const float* __restrict__ w1,
                  float* __restrict__ gated, float* __restrict__ cg) {
    __shared__ float As[MTILE * KP];
    __shared__ float Wu[NTILE * KP];
    __shared__ float Wv[NTILE * KP];

    const int t = threadIdx.x;
    const int wave = t >> 5, lane = t & 31;
    const int wm = wave & 3, wn = wave >> 2;   // 4x2 wave grid over 128x128
    const int ln = lane & 15, kh = lane >> 4;

    const int d0 = blockIdx.x * NTILE;
    const int m0 = blockIdx.y * MTILE;

    const float* xg = x + (size_t)m0 * D_MODEL;
    const float* wB = w1 + (size_t)(0 * D_MODEL + d0) * D_MODEL;  // Bg rows
    const float* wC = w1 + (size_t)(1 * D_MODEL + d0) * D_MODEL;  // Cg rows
    const float* wX = w1 + (size_t)(2 * D_MODEL + d0) * D_MODEL;  // Xg rows

    // ---- pass A: Bg and Xg in one sweep over K ----
    v8f accB[2][4] = {};
    v8f accX[2][4] = {};
    for (int k0 = 0; k0 < D_MODEL; k0 += KTILE) {
        load_tile_128x32(xg + k0, As, t);
        load_tile_128x32(wB + k0, Wu, t);
        load_tile_128x32(wX + k0, Wv, t);
        __syncthreads();
        wmma_ktile(As, Wu, accB, wm, wn, ln, kh);
        wmma_ktile(As, Wv, accX, wm, wn, ln, kh);
        __syncthreads();
    }
    // gated = Bg * Xg (lane/VGPR maps are identical for both accumulators)
#pragma unroll
    for (int i = 0; i < 2; ++i)
#pragma unroll
        for (int j = 0; j < 4; ++j)
#pragma unroll
            for (int r = 0; r < 8; ++r) {
                const int row = m0 + wm * 32 + i * 16 + kh * 8 + r;
                const int col = d0 + wn * 64 + j * 16 + ln;
                gated[(size_t)row * D_MODEL + col] = accB[i][j][r] * accX[i][j][r];
            }

    // ---- pass B: Cg ----
    v8f accC[2][4] = {};
    for (int k0 = 0; k0 < D_MODEL; k0 += KTILE) {
        load_tile_128x32(xg + k0, As, t);
        load_tile_128x32(wC + k0, Wu, t);
        __syncthreads();
        wmma_ktile(As, Wu, accC, wm, wn, ln, kh);
        __syncthreads();
    }
    store_acc(cg, accC, m0, d0, wm, wn, ln, kh);
}

// ---------------------------------------------------------------------------
// Kernel 2: depthwise causal conv (K=4) along S + Cg multiply, float4 over d.
// In-place over cg buffer (each element is read only by its own writer).
// ---------------------------------------------------------------------------
__global__ __launch_bounds__(256)
void k2_conv(const float* __restrict__ gated, const float* __restrict__ cw,
             float* __restrict__ cg_h) {
    const int idx = blockIdx.x * 256 + threadIdx.x;   // over BS * D/4
    const int ND4 = D_MODEL / 4;                      // 512
    const int d4  = idx & (ND4 - 1);
    const int row = idx >> 9;                         // idx / 512
    const int s   = row & (SEQ - 1);                  // position within batch
    const int d   = d4 << 2;

    v4f sum = {0.f, 0.f, 0.f, 0.f};
#pragma unroll
    for (int k = 0; k < 4; ++k) {
        const int sp = s - 3 + k;                     // causal left pad = 3
        if (sp >= 0) {
            v4f g = *(const v4f*)(gated + (size_t)(row - 3 + k) * D_MODEL + d);
            sum.x += cw[(d + 0) * 4 + k] * g.x;
            sum.y += cw[(d + 1) * 4 + k] * g.y;
            sum.z += cw[(d + 2) * 4 + k] * g.z;
            sum.w += cw[(d + 3) * 4 + k] * g.w;
        }
    }
    v4f c = *(const v4f*)(cg_h + (size_t)row * D_MODEL + d);
    *(v4f*)(cg_h + (size_t)row * D_MODEL + d) = c * sum;
}

// ---------------------------------------------------------------------------
// Kernel 3: out = h @ w2^T  (same WMMA tile machinery)
// ---------------------------------------------------------------------------
__global__ __launch_bounds__(256)
void k3_gemm_out(const float* __restrict__ h, const float* __restrict__ w2,
                 float* __restrict__ out) {
    __shared__ float As[MTILE * KP];
    __shared__ float Ws[NTILE * KP];

    const int t = threadIdx.x;
    const int wave = t >> 5, lane = t & 31;
    const int wm = wave & 3, wn = wave >> 2;
    const int ln = lane & 15, kh = lane >> 4;

    const int n0 = blockIdx.x * NTILE;
    const int m0 = blockIdx.y * MTILE;

    const float* hg = h + (size_t)m0 * D_MODEL;
    const float* wg = w2 + (size_t)n0 * D_MODEL;

    v8f acc[2][4] = {};
    for (int k0 = 0; k0 < D_MODEL; k0 += KTILE) {
        load_tile_128x32(hg + k0, As, t);
        load_tile_128x32(wg + k0, Ws, t);
        __syncthreads();
        wmma_ktile(As, Ws, acc, wm, wn, ln, kh);
        __syncthreads();
    }
    store_acc(out, acc, m0, n0, wm, wn, ln, kh);
}

// ---------------------------------------------------------------------------
extern "C" void kernel_launch(void* const* d_in, const int* in_sizes, int n_in,
                              void* d_out, int out_size, void* d_ws, size_t ws_size,
                              hipStream_t stream) {
    const float* x  = (const float*)d_in[0];   // [4, 4096, 2048]
    const float* w1 = (const float*)d_in[1];   // [6144, 2048]
    const float* w2 = (const float*)d_in[2];   // [2048, 2048]
    const float* cw = (const float*)d_in[3];   // [2048, 1, 4]

    float* gated = (float*)d_ws;                       // [BS, D]
    float* cg    = gated + (size_t)BS * D_MODEL;       // [BS, D] -> becomes h in-place

    dim3 grid_gemm(D_MODEL / NTILE, BS / MTILE);       // (16, 128)
    k1_gemm_gate<<<grid_gemm, 256, 0, stream>>>(x, w1, gated, cg);

    const int n_conv_blocks = (BS * (D_MODEL / 4)) / 256;  // 32768
    k2_conv<<<n_conv_blocks, 256, 0, stream>>>(gated, cw, cg);

    k3_gemm_out<<<grid_gemm, 256, 0, stream>>>(cg, w2, (float*)d_out);
}